// SingleModule_64192581206610
// MI455X (gfx1250) — compile-verified
//
#include <hip/hip_runtime.h>
#include <hip/hip_bf16.h>

#define D 128
#define OUTW 124  // D - 4 after two VALID convs of width 3

typedef float v2f __attribute__((ext_vector_type(2)));
typedef float v8f __attribute__((ext_vector_type(8)));

// ---------------------------------------------------------------------------
// f32 atomic add -> global_atomic_add_f32 (L2-side atomic, no CAS loop)
// ---------------------------------------------------------------------------
__device__ __forceinline__ void atomAddF32(float* p, float v) {
    unsafeAtomicAdd(p, v);
}

// ---------------------------------------------------------------------------
// Degree: deg[i] = 1 (self loop) + sum_{e: col[e]==i} ew[e]; then dinv=rsqrt
// ---------------------------------------------------------------------------
__global__ void deg_init(float* __restrict__ deg, int n) {
    int i = blockIdx.x * blockDim.x + threadIdx.x;
    if (i < n) deg[i] = 1.0f;
}

__global__ void deg_accum(const long long* __restrict__ col,
                          const float* __restrict__ ew,
                          float* __restrict__ deg, int e_cnt) {
    int e = blockIdx.x * blockDim.x + threadIdx.x;
    if (e < e_cnt) atomAddF32(&deg[(int)col[e]], ew[e]);
}

__global__ void deg_finish(float* __restrict__ deg, int n) {
    int i = blockIdx.x * blockDim.x + threadIdx.x;
    if (i < n) deg[i] = rsqrtf(deg[i]);  // deg >= 1 always (self loop)
}

// ---------------------------------------------------------------------------
// C[N x 128] = A[N x 128] @ W[128 x 128] using V_WMMA_F32_16X16X4_F32.
// Block = 256 threads = 8 waves. Block owns a 16-row stripe; wave w owns
// columns [16w, 16w+16). A-tile (16x128) staged in LDS, shared by all waves.
// ---------------------------------------------------------------------------
__global__ void __launch_bounds__(256)
gemm_wmma_f32(const float* __restrict__ A, const float* __restrict__ W,
              float* __restrict__ C) {
    __shared__ float As[16][D + 4];

    const int m0  = blockIdx.x * 16;
    const int tid = threadIdx.x;

    // cooperative A-tile load: 2048 floats / 256 threads = 8 each, coalesced
    #pragma unroll
    for (int i = tid; i < 16 * D; i += 256) {
        int r = i >> 7, c = i & (D - 1);
        As[r][c] = A[(size_t)(m0 + r) * D + c];
    }
    __syncthreads();

    const int wave  = tid >> 5;
    const int lane  = tid & 31;
    const int n0    = wave * 16;
    const int lh    = lane & 15;
    const bool hi   = lane >= 16;

    v8f acc = {0.f, 0.f, 0.f, 0.f, 0.f, 0.f, 0.f, 0.f};

    #pragma unroll 4
    for (int k0 = 0; k0 < D; k0 += 4) {
        const int ka = k0 + (hi ? 2 : 0);
        // A fragment: 16x4, lane layout per ISA 7.12.2 (32-bit A 16x4)
        v2f a;
        a.x = As[lh][ka];
        a.y = As[lh][ka + 1];
        // B fragment: 4x16, rows striped across lanes within a VGPR
        v2f b;
        b.x = W[(size_t)ka * D + n0 + lh];
        b.y = W[(size_t)(ka + 1) * D + n0 + lh];
        acc = __builtin_amdgcn_wmma_f32_16x16x4_f32(
            /*neg_a=*/false, a, /*neg_b=*/false, b,
            /*c_mod=*/(short)0, acc, /*reuse_a=*/false, /*reuse_b=*/false);
    }

    // C/D layout: VGPR v -> M = v + (hi ? 8 : 0), N = lane%16
    const int colN  = n0 + lh;
    const int rbase = m0 + (hi ? 8 : 0);
    #pragma unroll
    for (int v = 0; v < 8; ++v)
        C[(size_t)(rbase + v) * D + colN] = acc[v];
}

// ---------------------------------------------------------------------------
// agg[i][:] = xw[i][:] * dinv[i]^2   (self-loop message, w=1, r=c=i)
// float4-vectorized: 32 float4 per row.
// ---------------------------------------------------------------------------
__global__ void agg_init4(const float4* __restrict__ xw,
                          const float* __restrict__ dinv,
                          float4* __restrict__ agg, int total4) {
    int idx = blockIdx.x * blockDim.x + threadIdx.x;
    if (idx < total4) {
        float di = dinv[idx >> 5];
        float s  = di * di;
        float4 v = xw[idx];
        v.x *= s; v.y *= s; v.z *= s; v.w *= s;
        agg[idx] = v;
    }
}

// ---------------------------------------------------------------------------
// One WAVE per edge: e is wave-uniform (scalarized index/weight loads),
// each lane handles 4 consecutive floats (global_load_b128 gather +
// 4 hardware f32 atomics). Everything L2-resident (192MB L2 holds both
// the 25.6MB gather source and 25.6MB accumulator).
// ---------------------------------------------------------------------------
__global__ void __launch_bounds__(256)
edge_scatter(const long long* __restrict__ row,
             const long long* __restrict__ col,
             const float* __restrict__ ew,
             const float* __restrict__ dinv,
             const float* __restrict__ xw,
             float* __restrict__ agg, int e_cnt) {
    int wave = __builtin_amdgcn_readfirstlane((int)(threadIdx.x >> 5));
    int e = blockIdx.x * 8 + wave;
    if (e >= e_cnt) return;
    int lane = threadIdx.x & 31;

    int r = (int)row[e];
    int c = (int)col[e];
    float nrm = dinv[r] * ew[e] * dinv[c];

    float4 v = *((const float4*)(xw + (size_t)r * D) + lane);
    float* dst = agg + (size_t)c * D + lane * 4;
    atomAddF32(dst + 0, v.x * nrm);
    atomAddF32(dst + 1, v.y * nrm);
    atomAddF32(dst + 2, v.z * nrm);
    atomAddF32(dst + 3, v.w * nrm);
}

// ---------------------------------------------------------------------------
// h = relu(agg + b)  (in place, float4)
// ---------------------------------------------------------------------------
__global__ void bias_relu4(float4* __restrict__ buf,
                           const float4* __restrict__ b, int total4) {
    int idx = blockIdx.x * blockDim.x + threadIdx.x;
    if (idx < total4) {
        float4 v  = buf[idx];
        float4 bb = b[idx & 31];
        v.x = fmaxf(v.x + bb.x, 0.0f);
        v.y = fmaxf(v.y + bb.y, 0.0f);
        v.z = fmaxf(v.z + bb.z, 0.0f);
        v.w = fmaxf(v.w + bb.w, 0.0f);
        buf[idx] = v;
    }
}

// ---------------------------------------------------------------------------
// Fused conv1(2x1x1x3)+relu -> conv2(1x2x1x3)+relu, per row of h.
// out[i][j] = relu( bc2 + sum_t sum_ch k2[ch][t] *
//                   relu( k1[ch][0]h[j+t] + k1[ch][1]h[j+t+1]
//                       + k1[ch][2]h[j+t+2] + bc1[ch] ) )
// ---------------------------------------------------------------------------
__global__ void __launch_bounds__(128)
conv_fused(const float* __restrict__ h, const float* __restrict__ k1,
           const float* __restrict__ bc1, const float* __restrict__ k2,
           const float* __restrict__ bc2, float* __restrict__ out, int n) {
    __shared__ float rowbuf[D];
    int i = blockIdx.x;
    int j = threadIdx.x;  // 0..127
    rowbuf[j] = h[(size_t)i * D + j];
    __syncthreads();
    if (j < OUTW) {
        float accv = bc2[0];
        #pragma unroll
        for (int t = 0; t < 3; ++t) {
            #pragma unroll
            for (int ch = 0; ch < 2; ++ch) {
                float c = fmaf(k1[ch * 3 + 0], rowbuf[j + t],
                          fmaf(k1[ch * 3 + 1], rowbuf[j + t + 1],
                          fmaf(k1[ch * 3 + 2], rowbuf[j + t + 2], bc1[ch])));
                c = fmaxf(c, 0.0f);
                accv = fmaf(k2[ch * 3 + t], c, accv);
            }
        }
        out[(size_t)i * OUTW + j] = fmaxf(accv, 0.0f);
    }
}

// ---------------------------------------------------------------------------
// Host-side launcher
// Inputs: 0:X 1:edge_index 2:edge_weight 3:W1 4:b1 5:W2 6:b2 7:k1 8:bc1 9:k2 10:bc2
// ---------------------------------------------------------------------------
extern "C" void kernel_launch(void* const* d_in, const int* in_sizes, int n_in,
                              void* d_out, int out_size, void* d_ws, size_t ws_size,
                              hipStream_t stream) {
    const float*     X   = (const float*)d_in[0];
    const long long* ei  = (const long long*)d_in[1];  // int64 (2, E)
    const float*     ew  = (const float*)d_in[2];
    const float*     W1  = (const float*)d_in[3];
    const float*     b1  = (const float*)d_in[4];
    const float*     W2  = (const float*)d_in[5];
    const float*     b2  = (const float*)d_in[6];
    const float*     k1  = (const float*)d_in[7];
    const float*     bc1 = (const float*)d_in[8];
    const float*     k2  = (const float*)d_in[9];
    const float*     bc2 = (const float*)d_in[10];
    float* out = (float*)d_out;

    const int N = in_sizes[0] / D;   // 50000
    const int E = in_sizes[2];       // 800000
    const long long* row = ei;       // edge_index[0]
    const long long* col = ei + E;   // edge_index[1]

    // workspace: bufA | bufB | dinv  (~51.4 MB)
    float* bufA = (float*)d_ws;                 // N*D
    float* bufB = bufA + (size_t)N * D;         // N*D
    float* dinv = bufB + (size_t)N * D;         // N

    const int total4 = N * D / 4;
    const int tb     = 256;
    const int gN     = (N + tb - 1) / tb;
    const int gE     = (E + tb - 1) / tb;
    const int gT4    = (total4 + tb - 1) / tb;
    const int gScat  = (E + 7) / 8;   // 8 waves (edges) per 256-thread block
    const int gGemm  = N / 16;        // 50000 % 16 == 0

    // normalization coefficients
    deg_init  <<<gN, tb, 0, stream>>>(dinv, N);
    deg_accum <<<gE, tb, 0, stream>>>(col, ew, dinv, E);
    deg_finish<<<gN, tb, 0, stream>>>(dinv, N);

    // ---- layer 1: h1 = relu(gcn(X, W1, b1)) -> bufB
    gemm_wmma_f32<<<gGemm, 256, 0, stream>>>(X, W1, bufA);
    agg_init4    <<<gT4, tb, 0, stream>>>((const float4*)bufA, dinv,
                                          (float4*)bufB, total4);
    edge_scatter <<<gScat, tb, 0, stream>>>(row, col, ew, dinv, bufA, bufB, E);
    bias_relu4   <<<gT4, tb, 0, stream>>>((float4*)bufB, (const float4*)b1,
                                          total4);

    // ---- layer 2: h2 = relu(gcn(h1, W2, b2)) -> bufB (ping-pong via bufA)
    gemm_wmma_f32<<<gGemm, 256, 0, stream>>>(bufB, W2, bufA);
    agg_init4    <<<gT4, tb, 0, stream>>>((const float4*)bufA, dinv,
                                          (float4*)bufB, total4);
    edge_scatter <<<gScat, tb, 0, stream>>>(row, col, ew, dinv, bufA, bufB, E);
    bias_relu4   <<<gT4, tb, 0, stream>>>((float4*)bufB, (const float4*)b2,
                                          total4);

    // ---- fused convs -> out (N x 124)
    conv_fused<<<N, 128, 0, stream>>>(bufB, k1, bc1, k2, bc2, out, N);
}